// IRONFORGEDiscovery_67912022884926
// MI455X (gfx1250) — compile-verified
//
#include <hip/hip_runtime.h>
#include <hip/hip_bf16.h>
#include <math.h>

// Problem constants (fixed by the reference)
#define Nn     4096
#define INDIM  45
#define HDIM   44
#define NHEAD  4
#define HEADD  11
#define NEDGE  131072
#define MASKW  128          // u32 words per row of the allow-bitmask

typedef __attribute__((ext_vector_type(2))) float v2f;
typedef __attribute__((ext_vector_type(8))) float v8f;

// ---------------------------------------------------------------------------
// Bitmask build: bit (i,j) set => attention from i to j is ALLOWED
// ---------------------------------------------------------------------------
__global__ __launch_bounds__(256) void zero_mask_kernel(unsigned* __restrict__ mask) {
    int i = blockIdx.x * blockDim.x + threadIdx.x;
    if (i < Nn * MASKW) mask[i] = 0u;
}

__global__ __launch_bounds__(256) void scatter_edges_kernel(const int* __restrict__ ei,
                                                            unsigned* __restrict__ mask) {
    int e = blockIdx.x * blockDim.x + threadIdx.x;
    if (e < NEDGE) {
        int r = ei[e];
        int c = ei[NEDGE + e];
        atomicOr(&mask[r * MASKW + (c >> 5)], 1u << (c & 31));
    }
    if (e < Nn) {  // self-loops always allowed
        atomicOr(&mask[e * MASKW + (e >> 5)], 1u << (e & 31));
    }
}

// ---------------------------------------------------------------------------
// Exact-fp32 GEMM via V_WMMA_F32_16X16X4_F32, fully unrolled over K.
// Y[Nn x 44] = X[Nn x KD] @ W[KD x 44] + bias, one wave per 16x16 tile.
// All loads are unconditional (clamped addresses + 0/1 mask multipliers) so
// EXEC stays all-1s throughout and the scheduler can hoist loads over WMMAs.
// Tiles: 256 row-tiles x 3 col-tiles = 768 waves = 96 blocks of 256 threads.
// ---------------------------------------------------------------------------
template <int KD>
__global__ __launch_bounds__(256)
void wmma_gemm_kernel(const float* __restrict__ X, const float* __restrict__ W,
                      const float* __restrict__ bias, float* __restrict__ Y) {
    int wave = (blockIdx.x * blockDim.x + threadIdx.x) >> 5;
    int lane = threadIdx.x & 31;
    int tm = (wave / 3) * 16;       // tile row base
    int tn = (wave % 3) * 16;       // tile col base
    int half = lane >> 4;           // 0: lanes 0-15, 1: lanes 16-31
    int lr = lane & 15;

    int arow = tm + lr;             // A row for this lane
    int bcol = tn + lr;             // B/D column for this lane
    bool cok = (bcol < HDIM);
    int bcolc = cok ? bcol : (HDIM - 1);   // clamped, always in-bounds
    float bmul = cok ? 1.0f : 0.0f;

    const float* __restrict__ Xr = X + (size_t)arow * KD;

    v8f c = {};
    constexpr int KITER = (KD + 3) / 4;
    #pragma unroll
    for (int ki = 0; ki < KITER; ++ki) {
        const int k0 = ki * 4;
        int ka = k0 + half * 2;     // K slots: lanes0-15 -> k0,k0+1 ; lanes16-31 -> k0+2,k0+3
        v2f a, b;
        if (k0 + 3 < KD) {
            // compile-time known: fully in range for both halves
            a.x = Xr[ka];
            a.y = Xr[ka + 1];
            b.x = W[ka * HDIM + bcolc] * bmul;
            b.y = W[(ka + 1) * HDIM + bcolc] * bmul;
        } else {
            // tail slice (only for KD==45): clamp addresses, mask values
            int   ka0 = (ka     < KD) ? ka       : (KD - 1);
            int   ka1 = (ka + 1 < KD) ? (ka + 1) : (KD - 1);
            float m0  = (ka     < KD) ? 1.0f : 0.0f;
            float m1  = (ka + 1 < KD) ? 1.0f : 0.0f;
            a.x = Xr[ka0] * m0;
            a.y = Xr[ka1] * m1;
            b.x = W[ka0 * HDIM + bcolc] * (bmul * m0);
            b.y = W[ka1 * HDIM + bcolc] * (bmul * m1);
        }
        c = __builtin_amdgcn_wmma_f32_16x16x4_f32(false, a, false, b,
                                                  (short)0, c, false, false);
    }
    if (cok) {
        float bv = bias[bcol];
        #pragma unroll
        for (int r = 0; r < 8; ++r) {
            // D layout: VGPR r -> M = r (lanes 0-15) / r+8 (lanes 16-31)
            Y[(size_t)(tm + half * 8 + r) * HDIM + bcol] = c[r] + bv;
        }
    }
}

// ---------------------------------------------------------------------------
// Sparse masked attention (exact: masked entries underflow to 0 in fp32).
// One wave per row; each lane scans 4 mask words; per-lane online softmax;
// cross-lane log-sum-exp butterfly merge via __shfl_xor.
// ---------------------------------------------------------------------------
__global__ __launch_bounds__(256)
void sparse_attn_kernel(const float* __restrict__ Q, const float* __restrict__ K,
                        const float* __restrict__ V, const float* __restrict__ T,
                        const unsigned* __restrict__ mask, float* __restrict__ O) {
    int row  = (blockIdx.x * blockDim.x + threadIdx.x) >> 5;
    int lane = threadIdx.x & 31;

    const float inv_scale  = 1.0f / 0.100001f;        // 1/(SCALE + 1e-6)
    const float inv_sqrt_d = 0.30151134457776363f;    // 1/sqrt(11)
    const float NEGBIG = -1e37f;

    float q[HDIM];
    #pragma unroll
    for (int c2 = 0; c2 < HDIM; ++c2) q[c2] = Q[row * HDIM + c2];

    float m[NHEAD], lsum[NHEAD], acc[HDIM];
    #pragma unroll
    for (int h = 0; h < NHEAD; ++h) { m[h] = -__builtin_inff(); lsum[h] = 0.0f; }
    #pragma unroll
    for (int c2 = 0; c2 < HDIM; ++c2) acc[c2] = 0.0f;

    const unsigned* mrow = mask + (size_t)row * MASKW;
    #pragma unroll
    for (int wi = 0; wi < MASKW / 32; ++wi) {
        int w = lane + 32 * wi;
        unsigned bits = mrow[w];
        int jbase = w * 32;
        while (bits) {
            int b = __ffs((int)bits) - 1;
            bits &= bits - 1u;
            int j = jbase + b;
            float tb = -__builtin_sqrtf(T[((size_t)row * Nn + j) * 2] * inv_scale);
            const float* Kj = K + (size_t)j * HDIM;
            const float* Vj = V + (size_t)j * HDIM;
            #pragma unroll
            for (int h = 0; h < NHEAD; ++h) {
                float s = 0.0f;
                #pragma unroll
                for (int d = 0; d < HEADD; ++d) s += q[h * HEADD + d] * Kj[h * HEADD + d];
                s = s * inv_sqrt_d + tb;
                float mn = fmaxf(m[h], s);
                float sc = (m[h] > NEGBIG) ? __expf(m[h] - mn) : 0.0f;
                float p  = __expf(s - mn);
                lsum[h] = lsum[h] * sc + p;
                #pragma unroll
                for (int d = 0; d < HEADD; ++d)
                    acc[h * HEADD + d] = acc[h * HEADD + d] * sc + p * Vj[h * HEADD + d];
                m[h] = mn;
            }
        }
    }

    // Cross-lane merge of (m, l, acc) — NaN-safe for empty lanes.
    #pragma unroll
    for (int off = 16; off >= 1; off >>= 1) {
        #pragma unroll
        for (int h = 0; h < NHEAD; ++h) {
            float mo = __shfl_xor(m[h], off, 32);
            float lo = __shfl_xor(lsum[h], off, 32);
            float mn = fmaxf(m[h], mo);
            float s0 = (m[h] > NEGBIG) ? __expf(m[h] - mn) : 0.0f;
            float s1 = (mo   > NEGBIG) ? __expf(mo   - mn) : 0.0f;
            lsum[h] = lsum[h] * s0 + lo * s1;
            #pragma unroll
            for (int d = 0; d < HEADD; ++d) {
                float ao = __shfl_xor(acc[h * HEADD + d], off, 32);
                acc[h * HEADD + d] = acc[h * HEADD + d] * s0 + ao * s1;
            }
            m[h] = mn;
        }
    }

    if (lane == 0) {
        #pragma unroll
        for (int h = 0; h < NHEAD; ++h) {
            float invl = (lsum[h] > 0.0f) ? (1.0f / lsum[h]) : 0.0f;
            #pragma unroll
            for (int d = 0; d < HEADD; ++d)
                O[(size_t)row * HDIM + h * HEADD + d] = acc[h * HEADD + d] * invl;
        }
    }
}

// ---------------------------------------------------------------------------
extern "C" void kernel_launch(void* const* d_in, const int* in_sizes, int n_in,
                              void* d_out, int out_size, void* d_ws, size_t ws_size,
                              hipStream_t stream) {
    const float* nf = (const float*)d_in[0];   // node_features [4096,45]
    const int*   ei = (const int*)  d_in[1];   // edge_index [2,131072]
    const float* td = (const float*)d_in[2];   // temporal_data [4096,4096,2]
    const float* Wp = (const float*)d_in[3];
    const float* bp = (const float*)d_in[4];
    const float* Wq = (const float*)d_in[5];
    const float* bq = (const float*)d_in[6];
    const float* Wk = (const float*)d_in[7];
    const float* bk = (const float*)d_in[8];
    const float* Wv = (const float*)d_in[9];
    const float* bv = (const float*)d_in[10];
    const float* Wo = (const float*)d_in[11];
    const float* bo = (const float*)d_in[12];
    float* out = (float*)d_out;

    char* ws = (char*)d_ws;
    unsigned* mask = (unsigned*)ws;                    // 4096*128*4 = 2 MB
    float* x   = (float*)(ws + (size_t)Nn * MASKW * 4);
    float* Qm  = x   + (size_t)Nn * HDIM;
    float* Km  = Qm  + (size_t)Nn * HDIM;
    float* Vm  = Km  + (size_t)Nn * HDIM;
    float* att = Vm  + (size_t)Nn * HDIM;

    zero_mask_kernel<<<(Nn * MASKW + 255) / 256, 256, 0, stream>>>(mask);
    scatter_edges_kernel<<<(NEDGE + 255) / 256, 256, 0, stream>>>(ei, mask);

    // 768 tiles -> 96 blocks x 8 waves
    wmma_gemm_kernel<INDIM><<<96, 256, 0, stream>>>(nf, Wp, bp, x);   // x = nf@Wp+bp
    wmma_gemm_kernel<HDIM> <<<96, 256, 0, stream>>>(x,  Wq, bq, Qm);
    wmma_gemm_kernel<HDIM> <<<96, 256, 0, stream>>>(x,  Wk, bk, Km);
    wmma_gemm_kernel<HDIM> <<<96, 256, 0, stream>>>(x,  Wv, bv, Vm);

    sparse_attn_kernel<<<Nn / 8, 256, 0, stream>>>(Qm, Km, Vm, td, mask, att);

    wmma_gemm_kernel<HDIM> <<<96, 256, 0, stream>>>(att, Wo, bo, out); // final proj
}